// Pool_layer_20847771254838
// MI455X (gfx1250) — compile-verified
//
#include <hip/hip_runtime.h>
#include <hip/hip_bf16.h>

typedef float v2f __attribute__((ext_vector_type(2)));
typedef float v8f __attribute__((ext_vector_type(8)));

#define BS 8
#define NV 4096
#define NC 128
#define DK 20
#define TS (NV / 4)   // 1024 samples per batch

__device__ __forceinline__ unsigned hash_u32(unsigned x) {
    x ^= x >> 17; x *= 0xed5ad4bbu;
    x ^= x >> 11; x *= 0xac4c1b51u;
    x ^= x >> 15; x *= 0x31848babu;
    x ^= x >> 14;
    return x;
}

// ---------------------------------------------------------------------------
// K1: fused Gram/WMMA distance + top-5 smallest per row (kNN-4 + self).
// One wave per 16-row tile. v_wmma_f32_16x16x4_f32 with K=3 padded to 4.
// All 32 lanes scan (8 columns each); branchless compare-exchange top-5;
// per-row 2-list merge at the end.
// ---------------------------------------------------------------------------
__global__ __launch_bounds__(32) void knn_wmma_kernel(const float* __restrict__ verts,
                                                      int* __restrict__ nbr_out) {
    __shared__ float s_tile[16][17];   // 17-stride: conflict-free on 64 banks
    __shared__ float s_qm[16];
    __shared__ float s_md[16][2][5];
    __shared__ int   s_mi[16][2][5];

    const int lane  = threadIdx.x;
    const int tiles = NV / 16;
    const int b     = blockIdx.x / tiles;
    const int m0    = (blockIdx.x % tiles) * 16;
    const float* Vb = verts + (size_t)b * NV * 3;

    // A-tile (16x4, K padded): lanes 0-15 hold K={0,1}, lanes 16-31 hold K={2,pad0}
    const int arow = m0 + (lane & 15);
    const float ax = Vb[arow * 3 + 0];
    const float ay = Vb[arow * 3 + 1];
    const float az = Vb[arow * 3 + 2];
    v2f A;
    if (lane < 16) { A.x = ax; A.y = ay; } else { A.x = az; A.y = 0.0f; }

    if (lane < 16) s_qm[lane] = ax * ax + ay * ay + az * az;
    __syncthreads();

    float qrow[8];
    const int rbase = (lane < 16) ? 0 : 8;
#pragma unroll
    for (int r = 0; r < 8; ++r) qrow[r] = s_qm[rbase + r];

    // every lane keeps the 5 smallest over its half-row (8 cols per tile)
    float bestd[5];
    int   besti[5];
#pragma unroll
    for (int i = 0; i < 5; ++i) { bestd[i] = __builtin_inff(); besti[i] = 0; }

    const int myrow = lane & 15;
    const int cbase = (lane < 16) ? 0 : 8;

    for (int n0 = 0; n0 < NV; n0 += 16) {
        if (n0 + 16 < NV) __builtin_prefetch(Vb + (size_t)(n0 + 16) * 3, 0, 3);

        const int col = n0 + (lane & 15);
        const float bx = Vb[col * 3 + 0];
        const float by = Vb[col * 3 + 1];
        const float bz = Vb[col * 3 + 2];
        const float qn = bx * bx + by * by + bz * bz;
        v2f B;
        if (lane < 16) { B.x = bx; B.y = by; } else { B.x = bz; B.y = 0.0f; }

        v8f Cacc = {0.f, 0.f, 0.f, 0.f, 0.f, 0.f, 0.f, 0.f};
        v8f D = __builtin_amdgcn_wmma_f32_16x16x4_f32(false, A, false, B,
                                                      (short)0, Cacc, false, false);
#pragma unroll
        for (int r = 0; r < 8; ++r) {
            const int mrow = m0 + rbase + r;
            float dist = fmaf(-2.0f, D[r], qrow[r] + qn);
            if (mrow == col) dist = -__builtin_inff();   // self always ranks first
            s_tile[rbase + r][lane & 15] = dist;
        }
        __syncthreads();

        // branchless keep-5-smallest: tail insert + 4-stage bubble (cndmask only)
#pragma unroll
        for (int j = 0; j < 8; ++j) {
            const int c    = cbase + j;
            const float dv = s_tile[myrow][c];
            const int  idx = n0 + c;
            const bool ins = dv < bestd[4];
            bestd[4] = ins ? dv  : bestd[4];
            besti[4] = ins ? idx : besti[4];
#pragma unroll
            for (int p = 4; p >= 1; --p) {
                const bool  sw = bestd[p] < bestd[p - 1];
                const float td = bestd[p - 1];
                const int   ti = besti[p - 1];
                bestd[p - 1] = sw ? bestd[p] : td;
                besti[p - 1] = sw ? besti[p] : ti;
                bestd[p]     = sw ? td : bestd[p];
                besti[p]     = sw ? ti : besti[p];
            }
        }
        __syncthreads();
    }

    // merge the two sorted half-row lists per row
    const int half = lane >> 4;
#pragma unroll
    for (int p = 0; p < 5; ++p) {
        s_md[myrow][half][p] = bestd[p];
        s_mi[myrow][half][p] = besti[p];
    }
    __syncthreads();

    if (lane < 16) {
        int i0 = 0, i1 = 0;
        int outi[5];
#pragma unroll
        for (int p = 0; p < 5; ++p) {
            const float d0 = s_md[lane][0][i0];
            const float d1 = s_md[lane][1][i1];
            const int   j0 = s_mi[lane][0][i0];
            const int   j1 = s_mi[lane][1][i1];
            const bool take0 = (d0 < d1) || ((d0 == d1) && (j0 <= j1));
            outi[p] = take0 ? j0 : j1;
            i0 += take0 ? 1 : 0;
            i1 += take0 ? 0 : 1;
        }
        const int row = m0 + lane;
        int* o = nbr_out + ((size_t)b * NV + row) * 4;
        o[0] = outi[1]; o[1] = outi[2]; o[2] = outi[3]; o[3] = outi[4];
    }
}

// ---------------------------------------------------------------------------
// K2: max-pool features over the 4 nearest neighbors. One block per vertex.
// ---------------------------------------------------------------------------
__global__ __launch_bounds__(128) void pool_feat_kernel(const float* __restrict__ fmap,
                                                        const int* __restrict__ nbr,
                                                        float* __restrict__ pooled) {
    const int row = blockIdx.x;          // over BS*NV
    const int c   = threadIdx.x;
    const int b   = row / NV;
    const float* fb = fmap + (size_t)b * NV * NC;
    const int* nb = nbr + (size_t)row * 4;
    float m = -__builtin_inff();
#pragma unroll
    for (int k = 0; k < 4; ++k)
        m = fmaxf(m, fb[(size_t)nb[k] * NC + c]);
    pooled[(size_t)row * NC + c] = m;
}

// ---------------------------------------------------------------------------
// K3: density sum over DK given neighbors.
// ---------------------------------------------------------------------------
__global__ __launch_bounds__(256) void density_kernel(const float* __restrict__ verts,
                                                      const int* __restrict__ nid,
                                                      float* __restrict__ s_out) {
    const int i = blockIdx.x * blockDim.x + threadIdx.x;   // over BS*NV
    if (i >= BS * NV) return;
    const int b = i / NV;
    const float* Vb = verts + (size_t)b * NV * 3;
    const float x = verts[(size_t)i * 3 + 0];
    const float y = verts[(size_t)i * 3 + 1];
    const float z = verts[(size_t)i * 3 + 2];
    float sum = 0.0f;
#pragma unroll 4
    for (int k = 0; k < DK; ++k) {
        const int j = nid[(size_t)i * DK + k];
        const float dx = x - Vb[j * 3 + 0];
        const float dy = y - Vb[j * 3 + 1];
        const float dz = z - Vb[j * 3 + 2];
        sum += sqrtf(dx * dx + dy * dy + dz * dz);
    }
    s_out[i] = sum;
}

// ---------------------------------------------------------------------------
// K4: per-batch min/max reduction of the density sums.
// ---------------------------------------------------------------------------
__global__ __launch_bounds__(256) void minmax_kernel(const float* __restrict__ s,
                                                     float* __restrict__ mnmx) {
    __shared__ float smn[256];
    __shared__ float smx[256];
    const int b = blockIdx.x;
    const int t = threadIdx.x;
    float mn = __builtin_inff(), mx = -__builtin_inff();
    for (int i = t; i < NV; i += 256) {
        const float v = s[(size_t)b * NV + i];
        mn = fminf(mn, v);
        mx = fmaxf(mx, v);
    }
    smn[t] = mn; smx[t] = mx;
    __syncthreads();
    for (int off = 128; off > 0; off >>= 1) {
        if (t < off) {
            smn[t] = fminf(smn[t], smn[t + off]);
            smx[t] = fmaxf(smx[t], smx[t + off]);
        }
        __syncthreads();
    }
    if (t == 0) { mnmx[b * 2] = smn[0]; mnmx[b * 2 + 1] = smx[0]; }
}

// ---------------------------------------------------------------------------
// K5: stratified sample. Bin, key=bin+u (bin0 -> 4+u), bitonic argsort in LDS,
// per-bin rank -> scatter selected indices. One block per batch.
// ---------------------------------------------------------------------------
__global__ __launch_bounds__(256) void sample_kernel(const float* __restrict__ s,
                                                     const float* __restrict__ mnmx,
                                                     int* __restrict__ sel) {
    __shared__ float skey[NV];
    __shared__ int   sidx[NV];
    __shared__ int   counts[3];
    __shared__ int   nsamp[3], cstart[3], sstart[3];

    const int b = blockIdx.x;
    const int t = threadIdx.x;
    const float mn  = mnmx[b * 2];
    const float mx  = mnmx[b * 2 + 1];
    const float inv = 1.0f / (mx - mn);

    if (t < 3) counts[t] = 0;
    __syncthreads();

    for (int i = t; i < NV; i += 256) {
        const float nd = (s[(size_t)b * NV + i] - mn) * inv;
        // searchsorted(left) over [0, 1/3, 2/3, 1]: count of edges strictly < nd
        const int bin = (nd > 0.0f) + (nd > (1.0f / 3.0f)) + (nd > (2.0f / 3.0f));
        const unsigned h = hash_u32((unsigned)(b * NV + i) * 0x9e3779b9u + 42u);
        const float u = (float)(h >> 8) * (1.0f / 16777216.0f);
        skey[i] = (bin == 0) ? (4.0f + u) : ((float)bin + u);
        sidx[i] = i;
        if (bin >= 1) atomicAdd(&counts[bin - 1], 1);
    }
    __syncthreads();

    // bitonic sort ascending over NV elements
    for (int k = 2; k <= NV; k <<= 1) {
        for (int j = k >> 1; j > 0; j >>= 1) {
            for (int i = t; i < NV; i += 256) {
                const int ixj = i ^ j;
                if (ixj > i) {
                    const bool up = ((i & k) == 0);
                    const float ki = skey[i], kj = skey[ixj];
                    if ((ki > kj) == up) {
                        skey[i] = kj; skey[ixj] = ki;
                        const int tmp = sidx[i]; sidx[i] = sidx[ixj]; sidx[ixj] = tmp;
                    }
                }
            }
            __syncthreads();
        }
    }

    if (t == 0) {
        const int tot = counts[0] + counts[1] + counts[2];
        int acc_c = 0, acc_s = 0;
        for (int q = 0; q < 3; ++q) {
            nsamp[q]  = (int)floorf((float)counts[q] / (float)tot * (float)TS);
            cstart[q] = acc_c; acc_c += counts[q];
            sstart[q] = acc_s; acc_s += nsamp[q];
        }
    }
    __syncthreads();

    for (int i = t; i < TS; i += 256) sel[(size_t)b * TS + i] = 0;
    __syncthreads();

    for (int p = t; p < NV; p += 256) {
        int sb = (int)skey[p];
        if (sb >= 4) sb = 0;                 // bin-0 entries sorted to the end
        if (sb >= 1) {
            const int bi   = sb - 1;
            const int rank = p - cstart[bi];
            if (rank < nsamp[bi])
                sel[(size_t)b * TS + sstart[bi] + rank] = sidx[p];
        }
    }
}

// ---------------------------------------------------------------------------
// K6: gather final outputs. One block per output sample.
// ---------------------------------------------------------------------------
__global__ __launch_bounds__(128) void gather_out_kernel(const float* __restrict__ verts,
                                                         const float* __restrict__ pooled,
                                                         const int* __restrict__ sel,
                                                         float* __restrict__ out_v,
                                                         float* __restrict__ out_f) {
    const int p = blockIdx.x;            // over BS*TS
    const int t = threadIdx.x;
    const int b = p / TS;
    const int src = sel[p];
    const size_t srow = (size_t)b * NV + src;
    if (t < 3)
        out_v[(size_t)p * 3 + t] = verts[srow * 3 + t];
    out_f[(size_t)p * NC + t] = pooled[srow * NC + t];
}

extern "C" void kernel_launch(void* const* d_in, const int* in_sizes, int n_in,
                              void* d_out, int out_size, void* d_ws, size_t ws_size,
                              hipStream_t stream) {
    const float* verts = (const float*)d_in[0];
    const float* fmap  = (const float*)d_in[1];
    const int*   nid   = (const int*)d_in[2];

    char* ws = (char*)d_ws;
    size_t off = 0;
    int*   ws_nbr    = (int*)(ws + off);   off += (size_t)BS * NV * 4 * sizeof(int);
    float* ws_pooled = (float*)(ws + off); off += (size_t)BS * NV * NC * sizeof(float);
    float* ws_s      = (float*)(ws + off); off += (size_t)BS * NV * sizeof(float);
    float* ws_mnmx   = (float*)(ws + off); off += 256;
    int*   ws_sel    = (int*)(ws + off);   off += (size_t)BS * TS * sizeof(int);

    float* out_v = (float*)d_out;
    float* out_f = out_v + (size_t)BS * TS * 3;

    knn_wmma_kernel<<<BS * (NV / 16), 32, 0, stream>>>(verts, ws_nbr);
    pool_feat_kernel<<<BS * NV, NC, 0, stream>>>(fmap, ws_nbr, ws_pooled);
    density_kernel<<<(BS * NV + 255) / 256, 256, 0, stream>>>(verts, nid, ws_s);
    minmax_kernel<<<BS, 256, 0, stream>>>(ws_s, ws_mnmx);
    sample_kernel<<<BS, 256, 0, stream>>>(ws_s, ws_mnmx, ws_sel);
    gather_out_kernel<<<BS * TS, NC, 0, stream>>>(verts, ws_pooled, ws_sel, out_v, out_f);
}